// lstm_10694468567174
// MI455X (gfx1250) — compile-verified
//
#include <hip/hip_runtime.h>

// ---------------------------------------------------------------------------
// SRU stack (Lei et al. 2017) for MI455X / gfx1250.
//
// Shapes: B=40, T=20, FEA=200, H=1024, OUT=1095, L=12.
// ~64 GFLOP total; weights (~160MB) are L2-resident -> compute bound on the
// matrix path. fp32 reference + 12-layer recurrence => use exact-fp32
// V_WMMA_F32_16X16X4_F32 (not bf16).
//
// GEMM: 64x128 block tile, K chunks of 32 through double-buffered LDS.
//  - A staged row-major (pitch 36 floats): A fragments = one ds_load_b64.
//  - B staged k-pair-interleaved (float2 per [kpair][n], pitch 144 pairs):
//    B fragments = one ds_load_b64, no packing movs; lane-halves hit
//    disjoint LDS bank halves (2*144 dwords % 64 banks == 32).
//  - Global loads for chunk ch+1 staged in registers while computing ch;
//    fragment loads software-pipelined across k-steps.
// ---------------------------------------------------------------------------

typedef __attribute__((ext_vector_type(2))) float v2f;
typedef __attribute__((ext_vector_type(4))) float v4f;
typedef __attribute__((ext_vector_type(8))) float v8f;

#define BATCH   40
#define TSTEPS  20
#define FEA_DIM 200
#define HDIM    1024
#define OUTDIM  1095
#define LAYERS  12
#define MROWS   (BATCH * TSTEPS)   // 800

#define TM  64
#define TN  128
#define TK  32
#define LDA 36        // A pitch in floats (row 144B: 16B aligned, conflict-free)
#define PITCH2 144    // B pitch in float2 per k-pair row (2*144 % 64 == 32)
#define BPAIRS (TK / 2)
#define SA_ELEMS (TM * LDA)
#define SB_ELEMS (BPAIRS * PITCH2 * 2)

// ---- global -> register staging (with M/K/N edge guards) ----
__device__ __forceinline__ void load_a_regs(const float* Ar /*null if row OOB*/,
                                            int K, int k0, int aCol, v4f a[2])
{
#pragma unroll
    for (int u = 0; u < 2; ++u) {
        const int gk = k0 + aCol + u * 4;
        v4f v = {0.f, 0.f, 0.f, 0.f};
        if (Ar) {
            if (((K & 3) == 0) && (gk + 4 <= K)) {
                v = *(const v4f*)(Ar + gk);
            } else {
#pragma unroll
                for (int e = 0; e < 4; ++e)
                    if (gk + e < K) v[e] = Ar[gk + e];
            }
        }
        a[u] = v;
    }
}

__device__ __forceinline__ void load_b_regs(const float* __restrict__ W,
                                            int K, int N, int gk, int n0,
                                            int bLane, v4f b[4])
{
    const bool kOK = gk < K;
    const float* Wr = W + (size_t)gk * N;
#pragma unroll
    for (int j = 0; j < 4; ++j) {
        const int nc = (bLane + 8 * j) * 4;
        const int gn = n0 + nc;
        v4f v = {0.f, 0.f, 0.f, 0.f};
        if (kOK) {
            if (((N & 3) == 0) && (gn + 4 <= N)) {
                v = *(const v4f*)(Wr + gn);
            } else {
#pragma unroll
                for (int e = 0; e < 4; ++e)
                    if (gn + e < N) v[e] = Wr[gn + e];
            }
        }
        b[j] = v;
    }
}

// ---- register -> LDS staging ----
__device__ __forceinline__ void store_a_lds(float* sA, int aRow, int aCol,
                                            const v4f a[2])
{
#pragma unroll
    for (int u = 0; u < 2; ++u)
        *(v4f*)(sA + aRow * LDA + aCol + u * 4) = a[u];
}

__device__ __forceinline__ void store_b_lds(float* sB, int bRow, int bLane,
                                            const v4f b[4])
{
    const int p   = bRow >> 1;
    const int sub = bRow & 1;
    float* base = sB + p * (PITCH2 * 2) + sub;
#pragma unroll
    for (int j = 0; j < 4; ++j) {
        const int nc = (bLane + 8 * j) * 4;
#pragma unroll
        for (int e = 0; e < 4; ++e)
            base[(nc + e) * 2] = b[j][e];
    }
}

// ---- WMMA compute over one staged chunk (software-pipelined k-steps) ----
__device__ __forceinline__ void compute_chunk(const float* __restrict__ sA,
                                              const float* __restrict__ sB,
                                              int wm, int wn, int lh, int ln,
                                              v8f acc[4])
{
    const float* Abase = sA + (wm * 16 + ln) * LDA;
    const v2f*   Bv    = (const v2f*)sB;
    const int    nb    = wn * 64 + ln;

    // preload k-step 0 fragments (k-pair index = kk/2 + lane-half)
    v2f a = *(const v2f*)(Abase + 2 * lh);
    v2f b[4];
#pragma unroll
    for (int j = 0; j < 4; ++j) b[j] = Bv[lh * PITCH2 + nb + j * 16];

#pragma unroll
    for (int kk = 0; kk < TK; kk += 4) {
        v2f a_n = a;
        v2f b_n[4];
#pragma unroll
        for (int j = 0; j < 4; ++j) b_n[j] = b[j];
        if (kk + 4 < TK) {
            const int kp = (kk + 4) / 2 + lh;
            a_n = *(const v2f*)(Abase + 2 * kp);
#pragma unroll
            for (int j = 0; j < 4; ++j) b_n[j] = Bv[kp * PITCH2 + nb + j * 16];
        }
#pragma unroll
        for (int j = 0; j < 4; ++j) {
            acc[j] = __builtin_amdgcn_wmma_f32_16x16x4_f32(
                false, a, false, b[j], (short)0, acc[j], false, false);
        }
        a = a_n;
#pragma unroll
        for (int j = 0; j < 4; ++j) b[j] = b_n[j];
    }
}

// C[M,N] = EPI( A[M,K] @ W[K,N] + bias )
// IN_MAP : A-row for logical row r is (r%T)*B + r/T   (read time-major h)
// OUT_MAP: C-row for logical row r is (r%T)*B + r/T   (write time-major h)
// EPI    : 0 = none, 1 = relu
template <int IN_MAP, int OUT_MAP, int EPI>
__global__ __launch_bounds__(256)
void gemm_wmma_f32(const float* __restrict__ A, const float* __restrict__ W,
                   const float* __restrict__ bias, float* __restrict__ C,
                   int M, int N, int K)
{
    __shared__ __align__(16) float sA[2 * SA_ELEMS];
    __shared__ __align__(16) float sB[2 * SB_ELEMS];

    const int tid  = threadIdx.x;
    const int lane = tid & 31;         // wave32
    const int wv   = tid >> 5;         // 0..7
    const int wm   = wv & 3;           // M subtile (16 rows)
    const int wn   = wv >> 2;          // N half (64 cols)
    const int lh   = lane >> 4;        // lane half
    const int ln   = lane & 15;

    const int m0 = blockIdx.y * TM;
    const int n0 = blockIdx.x * TN;

    v8f acc[4];
#pragma unroll
    for (int j = 0; j < 4; ++j)
#pragma unroll
        for (int i = 0; i < 8; ++i) acc[j][i] = 0.0f;

    // staging thread assignment
    const int aRow  = tid >> 2;        // 0..63
    const int aCol  = (tid & 3) * 8;   // 0,8,16,24
    const int bRow  = tid >> 3;        // 0..31
    const int bLane = tid & 7;

    // A source row (mapping independent of k-chunk: hoist)
    const int  gr  = m0 + aRow;
    const bool rOK = gr < M;
    int src = gr;
    if (IN_MAP && rOK) src = (gr % TSTEPS) * BATCH + (gr / TSTEPS);
    const float* Ar = rOK ? (A + (size_t)src * K) : (const float*)nullptr;

    const int nChunks = (K + TK - 1) / TK;

    // prologue: stage chunk 0
    v4f aReg[2], bReg[4];
    load_a_regs(Ar, K, 0, aCol, aReg);
    load_b_regs(W, K, N, bRow, n0, bLane, bReg);
    store_a_lds(sA, aRow, aCol, aReg);
    store_b_lds(sB, bRow, bLane, bReg);
    __syncthreads();

    int buf = 0;
    for (int ch = 0; ch < nChunks; ++ch) {
        const bool more = (ch + 1 < nChunks);
        if (more) {
            const int k0n = (ch + 1) * TK;
            load_a_regs(Ar, K, k0n, aCol, aReg);
            load_b_regs(W, K, N, k0n + bRow, n0, bLane, bReg);
        }
        compute_chunk(sA + buf * SA_ELEMS, sB + buf * SB_ELEMS,
                      wm, wn, lh, ln, acc);
        if (more) {
            store_a_lds(sA + (buf ^ 1) * SA_ELEMS, aRow, aCol, aReg);
            store_b_lds(sB + (buf ^ 1) * SB_ELEMS, bRow, bLane, bReg);
            __syncthreads();
            buf ^= 1;
        }
    }

    // epilogue: C/D layout c[i] = C[m0 + wm*16 + i + 8*lh][n]
#pragma unroll
    for (int j = 0; j < 4; ++j) {
        const int gn = n0 + wn * 64 + j * 16 + ln;
        if (gn >= N) continue;
        const float bv = bias ? bias[gn] : 0.0f;
#pragma unroll
        for (int i = 0; i < 8; ++i) {
            const int grow = m0 + wm * 16 + i + 8 * lh;
            if (grow >= M) continue;
            float v = acc[j][i] + bv;
            if (EPI == 1) v = fmaxf(v, 0.0f);
            const int orow = OUT_MAP ? (grow % TSTEPS) * BATCH + (grow / TSTEPS)
                                     : grow;
            C[(size_t)orow * N + gn] = v;
        }
    }
}

// ---------------------------------------------------------------------------
// SRU recurrence. One thread per (b, h): 40960 independent length-20 scans.
// U is [T*B, 3H] time-major ([x_tilde | uf | ur]); h is [T*B, H] time-major
// updated in place (each element touched by exactly one thread).
// ---------------------------------------------------------------------------
__global__ __launch_bounds__(256)
void sru_scan(const float* __restrict__ U, const float* __restrict__ bvec,
              const float* __restrict__ c0, float* __restrict__ h,
              float* __restrict__ c_out)
{
    const int idx = blockIdx.x * blockDim.x + threadIdx.x;
    if (idx >= BATCH * HDIM) return;
    const int b  = idx / HDIM;
    const int hh = idx - b * HDIM;

    const float bf = bvec[hh];
    const float br = bvec[HDIM + hh];
    float c = c0[idx];

#pragma unroll 4
    for (int t = 0; t < TSTEPS; ++t) {
        const int row = t * BATCH + b;
        const float* Ur = U + (size_t)row * (3 * HDIM);
        const float xt = Ur[hh];
        const float f  = 1.0f / (1.0f + __expf(-(Ur[HDIM + hh] + bf)));
        const float r  = 1.0f / (1.0f + __expf(-(Ur[2 * HDIM + hh] + br)));
        c = f * c + (1.0f - f) * xt;
        float* hp = h + (size_t)row * HDIM + hh;
        const float xin = *hp;
        *hp = r * tanhf(c) + (1.0f - r) * xin;
    }
    c_out[idx] = c;
}

// ---------------------------------------------------------------------------
// Launch: dense1 -> 12 x (GEMM + scan) -> W3+relu -> W4.
// d_out = [800*1095 logits | 12*40*1024 final cell states].
// ---------------------------------------------------------------------------
extern "C" void kernel_launch(void* const* d_in, const int* in_sizes, int n_in,
                              void* d_out, int out_size, void* d_ws, size_t ws_size,
                              hipStream_t stream)
{
    (void)in_sizes; (void)n_in; (void)out_size; (void)ws_size;

    const float* x    = (const float*)d_in[0];  // [B,T,FEA]
    const float* hid0 = (const float*)d_in[1];  // [L,B,H]
    const float* W1   = (const float*)d_in[2];  // [FEA,H]
    const float* b1   = (const float*)d_in[3];  // [H]
    const float* sruW = (const float*)d_in[4];  // [L,H,3H]
    const float* sruB = (const float*)d_in[5];  // [L,2H]
    const float* W3   = (const float*)d_in[6];  // [H,H]
    const float* b3   = (const float*)d_in[7];  // [H]
    const float* W4   = (const float*)d_in[8];  // [H,OUT]
    const float* b4   = (const float*)d_in[9];  // [OUT]

    float* out     = (float*)d_out;                          // [800, OUT]
    float* hid_out = out + (size_t)MROWS * OUTDIM;           // [L,B,H]

    float* h  = (float*)d_ws;                                // [T*B, H] time-major
    float* U  = h + (size_t)MROWS * HDIM;                    // [T*B, 3H]
    float* h3 = U + (size_t)MROWS * 3 * HDIM;                // [B*T, H] batch-major

    const dim3 blk(256);
    const int mGrid = (MROWS + TM - 1) / TM;                 // 13

    // Dense layer 1: x(bt-major) @ W1 + b1 -> h (tb-major via OUT_MAP)
    gemm_wmma_f32<0, 1, 0><<<dim3((HDIM + TN - 1) / TN, mGrid), blk, 0, stream>>>(
        x, W1, b1, h, MROWS, HDIM, FEA_DIM);

    // SRU stack
    for (int l = 0; l < LAYERS; ++l) {
        gemm_wmma_f32<0, 0, 0><<<dim3((3 * HDIM + TN - 1) / TN, mGrid), blk, 0, stream>>>(
            h, sruW + (size_t)l * HDIM * 3 * HDIM, (const float*)nullptr, U,
            MROWS, 3 * HDIM, HDIM);
        sru_scan<<<(BATCH * HDIM + 255) / 256, blk, 0, stream>>>(
            U, sruB + (size_t)l * 2 * HDIM, hid0 + (size_t)l * BATCH * HDIM,
            h, hid_out + (size_t)l * BATCH * HDIM);
    }

    // W3 + relu: read h (tb-major) in bt order via IN_MAP -> h3 (bt-major)
    gemm_wmma_f32<1, 0, 1><<<dim3((HDIM + TN - 1) / TN, mGrid), blk, 0, stream>>>(
        h, W3, b3, h3, MROWS, HDIM, HDIM);

    // W4: h3 @ W4 + b4 -> logits
    gemm_wmma_f32<0, 0, 0><<<dim3((OUTDIM + TN - 1) / TN, mGrid), blk, 0, stream>>>(
        h3, W4, b4, out, MROWS, OUTDIM, HDIM);
}